// Message3Passing_30803505447332
// MI455X (gfx1250) — compile-verified
//
#include <hip/hip_runtime.h>
#include <hip/hip_bf16.h>
#include <stdint.h>

typedef unsigned int u32;
typedef unsigned long long u64;

typedef u32 u32x4 __attribute__((ext_vector_type(4)));
typedef int i32x4 __attribute__((ext_vector_type(4)));
typedef int i32x8 __attribute__((ext_vector_type(8)));

#define TDM_TILE_ROWS 1024
#define WAVES_PER_BLOCK 8

// ---------------------------------------------------------------------------
// TDM: async-stage a strided 2D tile of 32-bit words into LDS.
// Copies `rows` rows of `tile_dw` dwords, row pitch `row_stride_dw` dwords,
// from `gptr` into LDS byte offset `lds_byte_off`. Rows past `rows_total`
// would read zero (we never consume them). D# packed per CDNA5 ISA §8.3/8.4.
// This toolchain's builtin takes 6 args: g0, g1, g2, g3, <v8i extra>, cpol.
// ---------------------------------------------------------------------------
__device__ __forceinline__ void tdm_load_tile_u32(
    u32 lds_byte_off, const void* gptr, u32 tile_dw, u32 rows,
    u32 rows_total, u32 row_stride_dw)
{
  u64 ga = (u64)(uintptr_t)gptr;
  u32x4 g0;
  g0.x = 1u;                                            // count=1 (valid), user
  g0.y = lds_byte_off;                                  // lds_addr
  g0.z = (u32)ga;                                       // global_addr[31:0]
  g0.w = (u32)((ga >> 32) & 0x01FFFFFFu) | (2u << 30);  // addr[56:32] | type=2
  i32x8 g1;
  g1[0] = (int)(2u << 16);                              // data_size = 4 bytes
  g1[1] = (int)((tile_dw & 0xFFFFu) << 16);             // tensor_dim0 lo16 @[63:48]
  g1[2] = (int)(((tile_dw >> 16) & 0xFFFFu) |           // tensor_dim0 hi16
                ((rows_total & 0xFFFFu) << 16));        // tensor_dim1 lo16
  g1[3] = (int)(((rows_total >> 16) & 0xFFFFu) |        // tensor_dim1 hi16
                ((tile_dw & 0xFFFFu) << 16));           // tile_dim0
  g1[4] = (int)(rows & 0xFFFFu);                        // tile_dim1 (tile_dim2=0)
  g1[5] = (int)row_stride_dw;                           // tensor_dim0_stride lo32
  g1[6] = 0;                                            // stride0 hi / stride1 lo
  g1[7] = 0;
  i32x4 g2; g2[0] = 1; g2[1] = 1; g2[2] = 0; g2[3] = 0; // tensor_dim2/3 = 1
  i32x4 g3; g3[0] = 0; g3[1] = (int)(1u << 16); g3[2] = 0; g3[3] = 0; // dim4=1
  i32x8 gx = {0, 0, 0, 0, 0, 0, 0, 0};                  // unused (gather idx blk)
  __builtin_amdgcn_tensor_load_to_lds(g0, g1, g2, g3, gx, 0);
}

// ---------------------------------------------------------------------------
__global__ void k_zero_u32(u32* p, int n) {
  int i = blockIdx.x * blockDim.x + threadIdx.x;
  if (i < n) p[i] = 0u;
}

// Pass 1: histogram of destination indices. TDM stages the index_i column
// (low dword of the int64 at record dword 4) into LDS, double-buffered.
__global__ void k_hist(const long long* __restrict__ idx, u32* __restrict__ counts,
                       int n_trip, int ntiles) {
  __shared__ u32 sbuf[2][TDM_TILE_ROWS];
  const u32 lds_base = (u32)(uintptr_t)(&sbuf[0][0]);

  if (threadIdx.x < 32 && (int)blockIdx.x < ntiles) {
    int rem = n_trip - (int)blockIdx.x * TDM_TILE_ROWS;
    int rows = rem < TDM_TILE_ROWS ? rem : TDM_TILE_ROWS;
    tdm_load_tile_u32(lds_base,
                      (const char*)idx + (u64)blockIdx.x * TDM_TILE_ROWS * 24 + 16,
                      1u, (u32)rows, (u32)rem, 6u);
  }
  int it = 0;
  for (int tile = blockIdx.x; tile < ntiles; tile += gridDim.x, ++it) {
    int cur = it & 1;
    int next = tile + gridDim.x;
    if (threadIdx.x < 32) {
      if (next < ntiles) {
        int rem = n_trip - next * TDM_TILE_ROWS;
        int rows = rem < TDM_TILE_ROWS ? rem : TDM_TILE_ROWS;
        tdm_load_tile_u32(lds_base + (u32)((cur ^ 1) * TDM_TILE_ROWS * 4),
                          (const char*)idx + (u64)next * TDM_TILE_ROWS * 24 + 16,
                          1u, (u32)rows, (u32)rem, 6u);
        __builtin_amdgcn_s_wait_tensorcnt(1);   // tile `tile` is complete
      } else {
        __builtin_amdgcn_s_wait_tensorcnt(0);
      }
    }
    __syncthreads();
    int gbase = tile * TDM_TILE_ROWS;
    for (int r = threadIdx.x; r < TDM_TILE_ROWS; r += blockDim.x) {
      if (gbase + r < n_trip) atomicAdd(&counts[sbuf[cur][r]], 1u);
    }
    __syncthreads();
  }
}

// Pass 2a: per-1024-chunk block sums.
__global__ void k_scan_a(const u32* __restrict__ counts, u32* __restrict__ partials, int n) {
  __shared__ u32 sd[256];
  int t = threadIdx.x;
  int base = blockIdx.x * 1024 + t * 4;
  u32 s = 0;
  for (int k = 0; k < 4; ++k) { int i = base + k; if (i < n) s += counts[i]; }
  sd[t] = s; __syncthreads();
  for (int off = 128; off > 0; off >>= 1) {
    if (t < off) sd[t] += sd[t + off];
    __syncthreads();
  }
  if (t == 0) partials[blockIdx.x] = sd[0];
}

// Pass 2b: exclusive scan of block sums (tiny), write total sentinel.
__global__ void k_scan_b(u32* partials, u32* offsets, int nb, int n_nodes) {
  if (blockIdx.x == 0 && threadIdx.x == 0) {
    u32 run = 0;
    for (int i = 0; i < nb; ++i) { u32 v = partials[i]; partials[i] = run; run += v; }
    offsets[n_nodes] = run;
  }
}

// Pass 2c: within-chunk exclusive scan -> offsets (and cursor copy).
__global__ void k_scan_c(const u32* __restrict__ counts, const u32* __restrict__ partials,
                         u32* __restrict__ offsets, u32* __restrict__ cursor, int n) {
  __shared__ u32 sd[256];
  int t = threadIdx.x;
  int base = blockIdx.x * 1024 + t * 4;
  u32 v[4]; u32 s = 0;
  for (int k = 0; k < 4; ++k) { int i = base + k; v[k] = (i < n) ? counts[i] : 0u; s += v[k]; }
  sd[t] = s; __syncthreads();
  for (int off = 1; off < 256; off <<= 1) {
    u32 add = (t >= off) ? sd[t - off] : 0u;
    __syncthreads();
    sd[t] += add;
    __syncthreads();
  }
  u32 run = sd[t] - s + partials[blockIdx.x];   // exclusive prefix for this thread
  for (int k = 0; k < 4; ++k) {
    int i = base + k;
    if (i < n) { offsets[i] = run; cursor[i] = run; run += v[k]; }
  }
}

// Pass 3: bucket-fill CSR. TDM stages dwords {j_lo, j_hi, i_lo} per record.
__global__ void k_fill(const long long* __restrict__ idx, u32* __restrict__ cursor,
                       u32* __restrict__ csr, int n_trip, int ntiles) {
  __shared__ u32 sbuf[2][TDM_TILE_ROWS * 3];
  const u32 lds_base = (u32)(uintptr_t)(&sbuf[0][0]);

  if (threadIdx.x < 32 && (int)blockIdx.x < ntiles) {
    int rem = n_trip - (int)blockIdx.x * TDM_TILE_ROWS;
    int rows = rem < TDM_TILE_ROWS ? rem : TDM_TILE_ROWS;
    tdm_load_tile_u32(lds_base,
                      (const char*)idx + (u64)blockIdx.x * TDM_TILE_ROWS * 24 + 8,
                      3u, (u32)rows, (u32)rem, 6u);
  }
  int it = 0;
  for (int tile = blockIdx.x; tile < ntiles; tile += gridDim.x, ++it) {
    int cur = it & 1;
    int next = tile + gridDim.x;
    if (threadIdx.x < 32) {
      if (next < ntiles) {
        int rem = n_trip - next * TDM_TILE_ROWS;
        int rows = rem < TDM_TILE_ROWS ? rem : TDM_TILE_ROWS;
        tdm_load_tile_u32(lds_base + (u32)((cur ^ 1) * TDM_TILE_ROWS * 3 * 4),
                          (const char*)idx + (u64)next * TDM_TILE_ROWS * 24 + 8,
                          3u, (u32)rows, (u32)rem, 6u);
        __builtin_amdgcn_s_wait_tensorcnt(1);
      } else {
        __builtin_amdgcn_s_wait_tensorcnt(0);
      }
    }
    __syncthreads();
    int gbase = tile * TDM_TILE_ROWS;
    for (int r = threadIdx.x; r < TDM_TILE_ROWS; r += blockDim.x) {
      if (gbase + r < n_trip) {
        u32 j = sbuf[cur][r * 3 + 0];
        u32 i = sbuf[cur][r * 3 + 2];
        u32 pos = atomicAdd(&cursor[i], 1u);
        csr[pos] = j;
      }
    }
    __syncthreads();
  }
}

// Pass 4: gather-side reduction. One wave32 per node; 32 lanes x float2 = one
// 64-wide row. Segment j-indices broadcast via readlane -> SGPR-base loads.
__global__ void k_gather(const float* __restrict__ x, const u32* __restrict__ offsets,
                         const u32* __restrict__ csr, float* __restrict__ out, int n_nodes) {
  int wid  = threadIdx.x >> 5;    // wave32
  int lane = threadIdx.x & 31;
  int node = blockIdx.x * WAVES_PER_BLOCK + wid;
  if (node >= n_nodes) return;
  u32 s = (u32)__builtin_amdgcn_readfirstlane((int)offsets[node]);
  u32 e = (u32)__builtin_amdgcn_readfirstlane((int)offsets[node + 1]);
  float ax = 0.f, ay = 0.f;
  for (u32 base = s; base < e; base += 32) {
    u32 n = e - base; if (n > 32) n = 32;
    u32 jv = ((u32)lane < n) ? csr[base + lane] : 0u;
    for (u32 c = 0; c < n; ++c) {
      u32 j = (u32)__builtin_amdgcn_readlane((int)jv, (int)c);
      const float2* row = (const float2*)(x + (u64)j * 64u);
      float2 v = row[lane];
      ax += v.x; ay += v.y;
    }
  }
  float2 r; r.x = ax; r.y = ay;
  ((float2*)(out + (u64)node * 64u))[lane] = r;
}

// ---------------- fallback (workspace too small): atomic scatter ------------
__global__ void k_zero_f32(float* p, long long n) {
  long long i = (long long)blockIdx.x * blockDim.x + threadIdx.x;
  if (i < n) p[i] = 0.f;
}

__global__ void k_scatter(const float* __restrict__ x, const long long* __restrict__ idx,
                          float* __restrict__ out, int n_trip) {
  int wid  = threadIdx.x >> 5;
  int lane = threadIdx.x & 31;
  int t = blockIdx.x * WAVES_PER_BLOCK + wid;
  if (t >= n_trip) return;
  u32 j = (u32)idx[(u64)t * 3 + 1];
  u32 i = (u32)idx[(u64)t * 3 + 2];
  float2 v = ((const float2*)(x + (u64)j * 64u))[lane];
  float* o = out + (u64)i * 64u + lane * 2;
  atomicAdd(o, v.x);
  atomicAdd(o + 1, v.y);
}

// ---------------------------------------------------------------------------
extern "C" void kernel_launch(void* const* d_in, const int* in_sizes, int n_in,
                              void* d_out, int out_size, void* d_ws, size_t ws_size,
                              hipStream_t stream) {
  const float* x        = (const float*)d_in[0];
  const long long* idx  = (const long long*)d_in[1];
  float* out            = (float*)d_out;
  const int D = 64;
  int n_nodes = in_sizes[0] / D;
  int n_trip  = in_sizes[1] / 3;

  int nb = (n_nodes + 1023) / 1024;
  size_t need = (size_t)4 * ((size_t)3 * n_nodes + 1 + (size_t)n_trip + (size_t)nb);

  if (ws_size >= need) {
    u32* counts   = (u32*)d_ws;
    u32* offsets  = counts + n_nodes;        // n_nodes + 1
    u32* cursor   = offsets + n_nodes + 1;
    u32* csr      = cursor + n_nodes;
    u32* partials = csr + n_trip;

    k_zero_u32<<<(n_nodes + 255) / 256, 256, 0, stream>>>(counts, n_nodes);

    int ntiles = (n_trip + TDM_TILE_ROWS - 1) / TDM_TILE_ROWS;
    int hgrid  = ntiles < 512 ? ntiles : 512;
    k_hist<<<hgrid, 256, 0, stream>>>(idx, counts, n_trip, ntiles);

    k_scan_a<<<nb, 256, 0, stream>>>(counts, partials, n_nodes);
    k_scan_b<<<1, 1, 0, stream>>>(partials, offsets, nb, n_nodes);
    k_scan_c<<<nb, 256, 0, stream>>>(counts, partials, offsets, cursor, n_nodes);

    k_fill<<<hgrid, 256, 0, stream>>>(idx, cursor, csr, n_trip, ntiles);

    k_gather<<<(n_nodes + WAVES_PER_BLOCK - 1) / WAVES_PER_BLOCK,
               32 * WAVES_PER_BLOCK, 0, stream>>>(x, offsets, csr, out, n_nodes);
  } else {
    long long total = (long long)n_nodes * D;
    k_zero_f32<<<(int)((total + 255) / 256), 256, 0, stream>>>(out, total);
    k_scatter<<<(n_trip + WAVES_PER_BLOCK - 1) / WAVES_PER_BLOCK, 256, 0, stream>>>(
        x, idx, out, n_trip);
  }
}